// CRF_23819888623991
// MI455X (gfx1250) — compile-verified
//
#include <hip/hip_runtime.h>
#include <cmath>

#define B_SZ 512
#define S_SZ 512
#define T_SZ 64
#define START_I 62
#define STOP_I 63
#define NEG_V -10000.0f

#define LOG2E 1.44269504088896340736f
#define LN2   0.69314718055994530942f

typedef __attribute__((ext_vector_type(2))) float v2f;
typedef __attribute__((ext_vector_type(8))) float v8f;

// ---------------------------------------------------------------------------
// Kernel 1: E = exp(transitions).  exp(-10000) underflows to exactly 0, so the
// START-row / STOP-column masking is preserved exactly in the matmul domain.
// ---------------------------------------------------------------------------
__global__ void crf_prep(const float* __restrict__ trans, float* __restrict__ Eexp) {
    int i = blockIdx.x * blockDim.x + threadIdx.x;
    if (i < T_SZ * T_SZ)
        Eexp[i] = __builtin_amdgcn_exp2f(trans[i] * LOG2E);
}

// ---------------------------------------------------------------------------
// Kernel 2: forward scan.  One workgroup = 16 batch rows, 4 waves.
// Wave w computes tag-tile [16w, 16w+16) for all 16 batch columns using
// 16 chained V_WMMA_F32_16X16X4_F32 (K = 64).
// All LDS arrays are batch-major so every LDS access is b64/b128 vectorized.
// feat tiles double-buffered into LDS via global_load_async_to_lds_b128.
// ---------------------------------------------------------------------------
__launch_bounds__(128)
__global__ void crf_scan(const float* __restrict__ feas,
                         const float* __restrict__ trans,
                         const int*   __restrict__ seq_len,
                         const float* __restrict__ Eexp,
                         float*       __restrict__ normOut) {
    constexpr int NB = 16;           // batch tile
    constexpr int TP = T_SZ + 4;     // LDS row pitch = 68 floats (16B aligned)

    __shared__ float alphaS[NB][TP];     // [batch][tag]
    __shared__ float pS[NB][TP];         // exp(alpha - rowmax), [batch][tag]
    __shared__ float featS[2][NB * TP];  // double-buffered feat tile
    __shared__ float rowmax[NB];
    __shared__ float tStop[T_SZ];
    __shared__ int   Ls[NB];

    const int tid  = threadIdx.x;
    const int wave = tid >> 5;
    const int lane = tid & 31;
    const int bb   = blockIdx.x * NB;

    // --- persistent A fragments: E rows [16*wave .. +15], K chunks of 4 ---
    // 32-bit A 16x4 layout: lanes0-15 {K0,K1}, lanes16-31 {K2,K3}, M = lane&15
    v2f A[16];
    {
        const int mrow = wave * 16 + (lane & 15);
        const int khi  = (lane >> 4) * 2;
        const float* erow = Eexp + mrow * T_SZ + khi;
        #pragma unroll
        for (int kc = 0; kc < 16; ++kc) {
            v2f a; a.x = erow[kc * 4]; a.y = erow[kc * 4 + 1];
            A[kc] = a;
        }
    }

    // --- alpha0 ---
    for (int idx = tid; idx < NB * T_SZ; idx += 128) {
        int b = idx >> 6, tg = idx & 63;
        alphaS[b][tg] = (tg == START_I) ? 0.0f : NEG_V;
    }
    if (tid < NB)   Ls[tid]    = seq_len[bb + tid];
    if (tid < T_SZ) tStop[tid] = trans[STOP_I * T_SZ + tid];

    // --- async feat staging: thread t owns 32B of the 16x64 tile ---
    const float* gbase = feas + (size_t)bb * S_SZ * T_SZ;
    const unsigned long long gbase_u = (unsigned long long)gbase;
    const int rr = tid >> 3;          // batch row within tile
    const int cc = (tid & 7) * 8;     // starting tag column
    const unsigned lds0 = (unsigned)(unsigned long long)&featS[0][0];
    const unsigned lds1 = (unsigned)(unsigned long long)&featS[1][0];
    const unsigned ldsoff_t = (unsigned)((rr * TP + cc) * 4);
    const unsigned goff_row = (unsigned)((rr * S_SZ * T_SZ + cc) * 4);

    {   // prefetch s = 0 into buffer 0
        unsigned ldsa = lds0 + ldsoff_t;
        unsigned goff = goff_row;
        asm volatile("global_load_async_to_lds_b128 %0, %1, %2 offset:0"
                     :: "v"(ldsa), "v"(goff), "s"(gbase_u) : "memory");
        asm volatile("global_load_async_to_lds_b128 %0, %1, %2 offset:16"
                     :: "v"(ldsa), "v"(goff), "s"(gbase_u) : "memory");
    }
    __syncthreads();

    // loop-invariant registers
    const int  n    = lane & 15;                 // batch column (N)
    const int  kb   = (lane >> 4) * 2;           // B-frag K sublane offset
    const int  mofs = wave * 16 + ((lane >> 4) ? 8 : 0);
    const int  Lreg = Ls[n];
    const float* frow0 = &featS[0][n * TP + mofs];
    const float* frow1 = &featS[1][n * TP + mofs];

    // fused max/p mapping: 8-lane group per batch row, 8 contiguous tags each
    const int bm = tid >> 3;
    const int j0 = (tid & 7) * 8;

    for (int s = 0; s < S_SZ; ++s) {
        const int buf = s & 1;

        // prefetch feat tile for step s+1 into the other buffer
        if (s + 1 < S_SZ) {
            unsigned ldsa = (buf ? lds0 : lds1) + ldsoff_t;
            unsigned goff = goff_row + (unsigned)((s + 1) * T_SZ * 4);
            asm volatile("global_load_async_to_lds_b128 %0, %1, %2 offset:0"
                         :: "v"(ldsa), "v"(goff), "s"(gbase_u) : "memory");
            asm volatile("global_load_async_to_lds_b128 %0, %1, %2 offset:16"
                         :: "v"(ldsa), "v"(goff), "s"(gbase_u) : "memory");
        }

        // --- fused rowmax + p: b128 LDS reads, shfl reduce, b128 stores ---
        {
            const float4 a0 = *(const float4*)&alphaS[bm][j0];
            const float4 a1 = *(const float4*)&alphaS[bm][j0 + 4];
            float m = fmaxf(fmaxf(fmaxf(a0.x, a0.y), fmaxf(a0.z, a0.w)),
                            fmaxf(fmaxf(a1.x, a1.y), fmaxf(a1.z, a1.w)));
            m = fmaxf(m, __shfl_xor(m, 1, 32));
            m = fmaxf(m, __shfl_xor(m, 2, 32));
            m = fmaxf(m, __shfl_xor(m, 4, 32));
            if ((tid & 7) == 0) rowmax[bm] = m;
            float4 p0, p1;
            p0.x = __builtin_amdgcn_exp2f((a0.x - m) * LOG2E);
            p0.y = __builtin_amdgcn_exp2f((a0.y - m) * LOG2E);
            p0.z = __builtin_amdgcn_exp2f((a0.z - m) * LOG2E);
            p0.w = __builtin_amdgcn_exp2f((a0.w - m) * LOG2E);
            p1.x = __builtin_amdgcn_exp2f((a1.x - m) * LOG2E);
            p1.y = __builtin_amdgcn_exp2f((a1.y - m) * LOG2E);
            p1.z = __builtin_amdgcn_exp2f((a1.z - m) * LOG2E);
            p1.w = __builtin_amdgcn_exp2f((a1.w - m) * LOG2E);
            *(float4*)&pS[bm][j0]     = p0;
            *(float4*)&pS[bm][j0 + 4] = p1;
        }

        // this step's feat tile was issued a full iteration ago; retire it,
        // then one barrier seals pS, rowmax and featS[buf] for everyone.
        if (s + 1 < S_SZ) {
            asm volatile("s_wait_asynccnt 2" ::: "memory");
        } else {
            asm volatile("s_wait_asynccnt 0" ::: "memory");
        }
        __syncthreads();

        // --- D = E_tile(16x64) x p(64x16): b64 B-frag loads + WMMA chain ---
        v2f bf[16];
        #pragma unroll
        for (int kc = 0; kc < 16; ++kc)
            bf[kc] = *(const v2f*)&pS[n][kc * 4 + kb];
        v8f d = {0.f, 0.f, 0.f, 0.f, 0.f, 0.f, 0.f, 0.f};
        #pragma unroll
        for (int kc = 0; kc < 16; ++kc) {
            d = __builtin_amdgcn_wmma_f32_16x16x4_f32(false, A[kc], false, bf[kc],
                                                      (short)0, d, false, false);
        }

        // --- alpha update: D layout lanes0-15 M=r, lanes16-31 M=r+8 ---
        if (s < Lreg) {
            const float rm    = rowmax[n];
            const float* frow = (buf ? frow1 : frow0);
            const float4 f0 = *(const float4*)&frow[0];
            const float4 f1 = *(const float4*)&frow[4];
            float4 o0, o1;
            o0.x = rm + __builtin_amdgcn_logf(d[0]) * LN2 + f0.x;
            o0.y = rm + __builtin_amdgcn_logf(d[1]) * LN2 + f0.y;
            o0.z = rm + __builtin_amdgcn_logf(d[2]) * LN2 + f0.z;
            o0.w = rm + __builtin_amdgcn_logf(d[3]) * LN2 + f0.w;
            o1.x = rm + __builtin_amdgcn_logf(d[4]) * LN2 + f1.x;
            o1.y = rm + __builtin_amdgcn_logf(d[5]) * LN2 + f1.y;
            o1.z = rm + __builtin_amdgcn_logf(d[6]) * LN2 + f1.z;
            o1.w = rm + __builtin_amdgcn_logf(d[7]) * LN2 + f1.w;
            *(float4*)&alphaS[n][mofs]     = o0;
            *(float4*)&alphaS[n][mofs + 4] = o1;
        }
        __syncthreads();
    }

    // --- norm[b] = logsumexp_i(alpha[b][i] + trans[STOP][i]) ---
    if (tid < NB) {
        const int b = tid;
        float m = -INFINITY;
        for (int i = 0; i < T_SZ; ++i) m = fmaxf(m, alphaS[b][i] + tStop[i]);
        float sum = 0.f;
        for (int i = 0; i < T_SZ; ++i)
            sum += __builtin_amdgcn_exp2f((alphaS[b][i] + tStop[i] - m) * LOG2E);
        normOut[bb + b] = m + __builtin_amdgcn_logf(sum) * LN2;
    }
}

// ---------------------------------------------------------------------------
// Kernel 3: gold-path scores + final loss.  One wave per batch row.
// ---------------------------------------------------------------------------
__launch_bounds__(128)
__global__ void crf_score(const float* __restrict__ feas,
                          const float* __restrict__ trans,
                          const int*   __restrict__ tag,
                          const int*   __restrict__ seq_len,
                          const float* __restrict__ norm,
                          float*       __restrict__ out) {
    const int wave = threadIdx.x >> 5;
    const int lane = threadIdx.x & 31;
    const int b = blockIdx.x * 4 + wave;
    if (b >= B_SZ) return;
    const int L = seq_len[b];
    const int* tg = tag + (size_t)b * S_SZ;
    const float* fb = feas + (size_t)b * S_SZ * T_SZ;
    float acc = 0.f;
    for (int k = lane; k <= S_SZ; k += 32) {
        if (k <= L) {                       // transition term k
            int prev = (k == 0) ? START_I : tg[k - 1];
            int next = (k < L) ? tg[k] : STOP_I;
            acc += trans[next * T_SZ + prev];
        }
        if (k < L) {                        // emission term k (L <= S)
            acc += fb[k * T_SZ + tg[k]];
        }
    }
    #pragma unroll
    for (int off = 16; off > 0; off >>= 1)
        acc += __shfl_xor(acc, off, 32);
    if (lane == 0) out[b] = norm[b] - acc;
}

// ---------------------------------------------------------------------------
extern "C" void kernel_launch(void* const* d_in, const int* in_sizes, int n_in,
                              void* d_out, int out_size, void* d_ws, size_t ws_size,
                              hipStream_t stream) {
    (void)in_sizes; (void)n_in; (void)out_size; (void)ws_size;
    const float* feas  = (const float*)d_in[0];
    const float* trans = (const float*)d_in[1];
    const int*   tagp  = (const int*)d_in[2];
    const int*   slen  = (const int*)d_in[3];
    float* out  = (float*)d_out;
    float* Eexp = (float*)d_ws;              // 64*64 floats
    float* norm = Eexp + T_SZ * T_SZ;        // 512 floats

    crf_prep <<<(T_SZ * T_SZ + 255) / 256, 256, 0, stream>>>(trans, Eexp);
    crf_scan <<<B_SZ / 16, 128, 0, stream>>>(feas, trans, slen, Eexp, norm);
    crf_score<<<B_SZ / 4, 128, 0, stream>>>(feas, trans, tagp, slen, norm, out);
}